// BatchedGaussianRenderer_34943853920759
// MI455X (gfx1250) — compile-verified
//
#include <hip/hip_runtime.h>
#include <hip/hip_bf16.h>

typedef float v2f __attribute__((ext_vector_type(2)));
typedef float v8f __attribute__((ext_vector_type(8)));

#define GN      4096
#define IMG_W   128
#define IMG_H   128
#define NPIX    (IMG_W * IMG_H)
#define EPSV    1e-6f

// ---------------------------------------------------------------------------
// Kernel 1: per-gaussian preprocessing. One thread per gaussian.
// Produces 8 polynomial coefficients per gaussian:
//   expo(x,y) = c0 + cx*x + cy*y + cxx*x^2 + cxy*x*y + cyy*y^2
// with the temporal log-weight folded into c0, so the splat contribution is
// exp(expo). Layout: coef[n*8 + k], k = {c0,cx,cy,cxx,cxy,cyy,0,0}.
// ---------------------------------------------------------------------------
__global__ void gauss_preprocess(const float* __restrict__ means,
                                 const float* __restrict__ raw_scales,
                                 const float* __restrict__ rotors,
                                 const float* __restrict__ t_ptr,
                                 const float* __restrict__ angle_ptr,
                                 float* __restrict__ coef) {
    int n = blockIdx.x * blockDim.x + threadIdx.x;
    if (n >= GN) return;

    const float t     = t_ptr[0];
    const float angle = angle_ptr[0];

    // scales^2
    float s2[4];
#pragma unroll
    for (int k = 0; k < 4; ++k) {
        float s = __expf(raw_scales[n * 4 + k]);
        s2[k] = s * s;
    }

    // quaternions
    const float* r = rotors + n * 8;
    float q1w = r[0], q1x = r[4], q1y = r[5], q1z = r[6];
    {
        float nn = sqrtf(q1w * q1w + q1x * q1x + q1y * q1y + q1z * q1z);
        float iv = 1.0f / fmaxf(nn, 1e-12f);
        q1w *= iv; q1x *= iv; q1y *= iv; q1z *= iv;
    }
    float q2w = r[7], q2x = -r[1], q2y = -r[2], q2z = -r[3];
    {
        float nn = sqrtf(q2w * q2w + q2x * q2x + q2y * q2y + q2z * q2z);
        float iv = 1.0f / fmaxf(nn, 1e-12f);
        q2w *= iv; q2x *= iv; q2y *= iv; q2z *= iv;
    }

    // left / right(conj) quaternion matrices
    const float Lm[4][4] = {
        { q1w, -q1x, -q1y, -q1z},
        { q1x,  q1w, -q1z,  q1y},
        { q1y,  q1z,  q1w, -q1x},
        { q1z, -q1y,  q1x,  q1w}};
    const float Rm[4][4] = {
        { q2w,  q2x,  q2y,  q2z},
        {-q2x,  q2w,  q2z, -q2y},
        {-q2y, -q2z,  q2w,  q2x},
        {-q2z,  q2y, -q2x,  q2w}};

    float R4[4][4];
#pragma unroll
    for (int i = 0; i < 4; ++i)
#pragma unroll
        for (int j = 0; j < 4; ++j) {
            float acc = 0.0f;
#pragma unroll
            for (int k = 0; k < 4; ++k) acc += Lm[i][k] * Rm[k][j];
            R4[i][j] = acc;
        }

    // cov4D = R4 diag(s2) R4^T
    float cov[4][4];
#pragma unroll
    for (int i = 0; i < 4; ++i)
#pragma unroll
        for (int k = 0; k < 4; ++k) {
            float acc = 0.0f;
#pragma unroll
            for (int j = 0; j < 4; ++j) acc += R4[i][j] * s2[j] * R4[k][j];
            cov[i][k] = acc;
        }

    const float Wm  = fmaxf(cov[3][3], EPSV);
    const float lam = 1.0f / Wm;
    const float dt  = t - means[n * 4 + 3];
    const float V0 = cov[0][3], V1 = cov[1][3], V2 = cov[2][3];

    const float mu0 = means[n * 4 + 0] + V0 * dt * lam;
    const float mu1 = means[n * 4 + 1] + V1 * dt * lam;
    const float mu2 = means[n * 4 + 2] + V2 * dt * lam;

    const float ca = __cosf(angle), sa = __sinf(angle);
    // view = [[ca,0,sa],[0,1,0],[-sa,0,ca]]
    const float mv0 = ca * mu0 + sa * mu2;
    const float mv1 = mu1;

    // cov3D = U - V V^T / Wm  (symmetric 3x3)
    const float M00 = cov[0][0] - V0 * V0 * lam;
    const float M01 = cov[0][1] - V0 * V1 * lam;
    const float M02 = cov[0][2] - V0 * V2 * lam;
    const float M11 = cov[1][1] - V1 * V1 * lam;
    const float M12 = cov[1][2] - V1 * V2 * lam;
    const float M22 = cov[2][2] - V2 * V2 * lam;

    // cov3D_view rows 0,1 (only entries needed for the 2x2 screen cov)
    const float t0 = ca * M00 + sa * M02;
    const float t1 = ca * M01 + sa * M12;
    const float t2 = ca * M02 + sa * M22;
    const float cv00 = ca * t0 + sa * t2;
    const float cv01 = t1;
    const float cv11 = M11;

    const float sxs = (IMG_W - 1) * 0.5f * 0.5f;  // (W-1)/2 * ZOOM = 31.75
    const float sys = (IMG_H - 1) * 0.5f * 0.5f;
    const float A2 = sxs * sxs * cv00 + EPSV;
    const float B2 = sxs * sys * cv01;
    const float D2 = sys * sys * cv11 + EPSV;
    const float idet = 1.0f / (A2 * D2 - B2 * B2);
    const float iA =  D2 * idet;
    const float iB = -B2 * idet;
    const float iD =  A2 * idet;

    const float mx = sxs * mv0 + (IMG_W - 1) * 0.5f;
    const float my = sys * mv1 + (IMG_H - 1) * 0.5f;

    const float logw = -0.5f * lam * dt * dt;
    const float c0  = -0.5f * (iA * mx * mx + 2.0f * iB * mx * my + iD * my * my) + logw;
    const float cx  = iA * mx + iB * my;
    const float cy  = iB * mx + iD * my;
    const float cxx = -0.5f * iA;
    const float cxy = -iB;
    const float cyy = -0.5f * iD;

    float* o = coef + (size_t)n * 8;
    o[0] = c0;  o[1] = cx;  o[2] = cy;  o[3] = cxx;
    o[4] = cxy; o[5] = cyy; o[6] = 0.0f; o[7] = 0.0f;
}

// ---------------------------------------------------------------------------
// Kernel 2: WMMA splat. Each wave32 owns a 16-pixel tile (M), streams all 4096
// gaussians in chunks of 16 (N), K=4+4 basis terms via two chained
// V_WMMA_F32_16X16X4_F32, then exp() + accumulate, finally a 16-lane-half
// shuffle reduction over the gaussian dimension of the D tile.
//
// A (16x4 f32) layout: lane m%16 = pixel row; VGPR0 = K{0|2}, VGPR1 = K{1|3}
//                      split by lane half.
// B (4x16 f32) layout: lane%16 = gaussian col; VGPR0 = K row {0|2},
//                      VGPR1 = K row {1|3} split by lane half.
// D (16x16 f32) layout: VGPR v = pixel (v + 8*(lane>=16)), lane%16 = gaussian.
// ---------------------------------------------------------------------------
__global__ void gauss_render(const float* __restrict__ coef,
                             float* __restrict__ img) {
    const int lane  = threadIdx.x & 31;
    const int wave  = threadIdx.x >> 5;
    const int tile  = blockIdx.x * (blockDim.x >> 5) + wave;   // 0..1023
    const int m     = lane & 15;
    const int hi    = lane >> 4;                               // lane half

    const int   p = tile * 16 + m;                             // flat pixel
    const float x = (float)(p & (IMG_W - 1));
    const float y = (float)(p >> 7);

    // Basis block 0: [1, x, y, x^2]   block 1: [x*y, y^2, 0, 0]
    v2f a1, a2;
    a1.x = hi ? y       : 1.0f;    // K = 0 or 2
    a1.y = hi ? x * x   : x;       // K = 1 or 3
    a2.x = hi ? 0.0f    : x * y;   // K = 4 or 6
    a2.y = hi ? 0.0f    : y * y;   // K = 5 or 7

    const int ncol = lane & 15;
    const int koff = hi ? 2 : 0;

    v8f acc = {0.f, 0.f, 0.f, 0.f, 0.f, 0.f, 0.f, 0.f};

    for (int g0 = 0; g0 < GN; g0 += 16) {
        const float* cb = coef + ((size_t)(g0 + ncol) * 8 + koff);
        const v2f b1 = *(const v2f*)(cb);       // coef K rows {koff, koff+1}
        const v2f b2 = *(const v2f*)(cb + 4);   // coef K rows {4+koff, 5+koff}

        v8f d = {0.f, 0.f, 0.f, 0.f, 0.f, 0.f, 0.f, 0.f};
        d = __builtin_amdgcn_wmma_f32_16x16x4_f32(false, a1, false, b1,
                                                  (short)0, d, false, false);
        d = __builtin_amdgcn_wmma_f32_16x16x4_f32(false, a2, false, b2,
                                                  (short)0, d, false, false);
#pragma unroll
        for (int i = 0; i < 8; ++i) acc[i] += __expf(d[i]);
    }

    // Reduce over the 16 gaussian columns (lanes within each 16-lane half).
#pragma unroll
    for (int i = 0; i < 8; ++i) {
        float v = acc[i];
        v += __shfl_xor(v, 1);
        v += __shfl_xor(v, 2);
        v += __shfl_xor(v, 4);
        v += __shfl_xor(v, 8);
        acc[i] = v;
    }

    // lane 0 holds pixel sums for rows 0..7, lane 16 for rows 8..15.
    if ((lane & 15) == 0) {
        const int base = tile * 16 + hi * 8;
#pragma unroll
        for (int i = 0; i < 8; ++i) img[base + i] = acc[i];
    }
}

// ---------------------------------------------------------------------------
// Kernel 3: single-block max reduction + normalization.
// ---------------------------------------------------------------------------
__global__ void gauss_finalize(const float* __restrict__ img,
                               float* __restrict__ out) {
    __shared__ float wmax[32];
    const int tid = threadIdx.x;   // blockDim.x == 1024 (32 waves)

    float mx = 0.0f;
    for (int i = tid; i < NPIX; i += 1024) mx = fmaxf(mx, img[i]);

#pragma unroll
    for (int off = 16; off > 0; off >>= 1) mx = fmaxf(mx, __shfl_xor(mx, off));
    if ((tid & 31) == 0) wmax[tid >> 5] = mx;
    __syncthreads();
    if (tid < 32) {
        float v = wmax[tid];
#pragma unroll
        for (int off = 16; off > 0; off >>= 1) v = fmaxf(v, __shfl_xor(v, off));
        if (tid == 0) wmax[0] = fmaxf(v, EPSV);
    }
    __syncthreads();

    const float inv = 1.0f / wmax[0];
    for (int i = tid; i < NPIX; i += 1024) out[i] = img[i] * inv;
}

// ---------------------------------------------------------------------------
extern "C" void kernel_launch(void* const* d_in, const int* in_sizes, int n_in,
                              void* d_out, int out_size, void* d_ws, size_t ws_size,
                              hipStream_t stream) {
    const float* means      = (const float*)d_in[0];
    const float* raw_scales = (const float*)d_in[1];
    const float* rotors     = (const float*)d_in[2];
    const float* t_ptr      = (const float*)d_in[3];
    const float* angle_ptr  = (const float*)d_in[4];
    float* out = (float*)d_out;

    float* coef = (float*)d_ws;          // 4096 * 8 floats = 128 KB
    float* img  = coef + (size_t)GN * 8; //   16384 floats  =  64 KB

    gauss_preprocess<<<GN / 256, 256, 0, stream>>>(means, raw_scales, rotors,
                                                   t_ptr, angle_ptr, coef);
    // 1024 wave-tiles of 16 pixels; 8 waves per block -> 128 blocks
    gauss_render<<<128, 256, 0, stream>>>(coef, img);
    gauss_finalize<<<1, 1024, 0, stream>>>(img, out);
}